// SoftPoint2Image_43516608643718
// MI455X (gfx1250) — compile-verified
//
#include <hip/hip_runtime.h>

#define RES   512
#define HWV   8      // half-width of kernel
#define NPTS_STRIDE 3

typedef __attribute__((ext_vector_type(2))) float v2f;
typedef __attribute__((ext_vector_type(8))) float v8f;

__global__ void SoftPoint2Image_zero_kernel(float4* __restrict__ out, int n4) {
    int i = blockIdx.x * blockDim.x + threadIdx.x;
    if (i < n4) out[i] = make_float4(0.0f, 0.0f, 0.0f, 0.0f);
}

// Broadcast a float from a fixed lane via v_readlane_b32 (no LDS / DScnt).
__device__ __forceinline__ float bcast_lane(float v, int lane) {
    return __uint_as_float(__builtin_amdgcn_readlane(__float_as_uint(v), lane));
}

// One point per wave32. WMMA f32 16x16x4 computes the 16x16 core of the
// rank-1 outer product w * gy(16x1) x gx(1x16); edge row/col (M=16 / N=16)
// handled by VALU with a lane-16 readlane broadcast. Scatter-add via
// L2-resident global_atomic_add_f32 (image is 1MB, fits in 192MB L2).
__global__ __launch_bounds__(256) void SoftPoint2Image_splat_kernel(
    const float* __restrict__ p, float* __restrict__ img, int npts) {
    const float inv      = 20000.0f;        // 1/(2*sigma^2), sigma=0.005
    const float linscale = 1.0f / 511.0f;   // linspace(0,1,512) spacing

    int tid    = blockIdx.x * blockDim.x + threadIdx.x;
    int wave   = tid >> 5;
    int lane   = tid & 31;
    int nwaves = (gridDim.x * blockDim.x) >> 5;

    for (int i = wave; i < npts; i += nwaves) {
        // prefetch next point this wave will touch (speculative, dropped if OOB)
        if (i + nwaves < npts) __builtin_prefetch(p + NPTS_STRIDE * (i + nwaves), 0, 0);

        // broadcast load: all lanes hit the same cacheline
        float cx = p[NPTS_STRIDE * i + 0];
        float cy = p[NPTS_STRIDE * i + 1];
        float w  = p[NPTS_STRIDE * i + 2];
        int ccx = (int)floorf(cx * (float)RES);
        int ccy = (int)floorf(cy * (float)RES);

        // lanes 0..16 evaluate the separable Gaussian taps (mask folded in)
        int col  = ccx - HWV + lane;
        int row  = ccy - HWV + lane;
        int colc = min(max(col, 0), RES - 1);
        int rowc = min(max(row, 0), RES - 1);
        float ddx = (float)colc * linscale - cx;
        float ddy = (float)rowc * linscale - cy;
        float ex  = __expf(-ddx * ddx * inv);
        float ey  = __expf(-ddy * ddy * inv);
        float gxl = (lane <= 16 && col >= 0 && col < RES) ? ex : 0.0f;
        float gyl = (lane <= 16 && row >= 0 && row < RES) ? ey : 0.0f;

        // A (16x4 f32): lanes 0-15 hold rows M=0..15, VGPR0 = K=0 column.
        // Only K=0 column nonzero -> D = outer(w*gy[0:16], gx[0:16]).
        v2f a; a.x = (lane < 16) ? (w * gyl) : 0.0f; a.y = 0.0f;
        // B (4x16 f32): K=0 row in lanes 0-15 of VGPR0; everything else zero.
        v2f b; b.x = (lane < 16) ? gxl : 0.0f;       b.y = 0.0f;
        v8f c = {};
        v8f dm = __builtin_amdgcn_wmma_f32_16x16x4_f32(
            /*neg_a=*/false, a, /*neg_b=*/false, b,
            /*c_mod=*/(short)0, c, /*reuse_a=*/false, /*reuse_b=*/false);

        // D layout: VGPR v -> M = v + 8*(lane/16), N = lane%16
        int N     = lane & 15;
        int Mbase = (lane >> 4) << 3;
        int cc0   = min(max(ccx - HWV + N, 0), RES - 1);
        int rbase = ccy - HWV + Mbase;
        #pragma unroll
        for (int v = 0; v < 8; ++v) {
            int rr = min(max(rbase + v, 0), RES - 1);
            atomicAdd(&img[rr * RES + cc0], dm[v]);
        }

        // edge taps: row M=16 (17 taps) and col N=16 (16 taps, corner already done)
        float gy16 = bcast_lane(gyl, 16);
        float gx16 = bcast_lane(gxl, 16);
        if (lane < 17) {
            int rr = min(max(ccy + HWV, 0), RES - 1);
            int cc = min(max(ccx - HWV + lane, 0), RES - 1);
            atomicAdd(&img[rr * RES + cc], w * gy16 * gxl);
        }
        if (lane < 16) {
            int rr = min(max(ccy - HWV + lane, 0), RES - 1);
            int cc = min(max(ccx + HWV, 0), RES - 1);
            atomicAdd(&img[rr * RES + cc], w * gyl * gx16);
        }
    }
}

extern "C" void kernel_launch(void* const* d_in, const int* in_sizes, int n_in,
                              void* d_out, int out_size, void* d_ws, size_t ws_size,
                              hipStream_t stream) {
    (void)n_in; (void)d_ws; (void)ws_size;
    const float* p   = (const float*)d_in[0];
    float*       out = (float*)d_out;
    int npts = in_sizes[0] / NPTS_STRIDE;   // 200000

    // zero all 4 channels with b128 stores (only channel 0 is written by the splat)
    int n4 = out_size / 4;                  // 4*512*512 / 4, out_size divisible by 4
    SoftPoint2Image_zero_kernel<<<(n4 + 255) / 256, 256, 0, stream>>>((float4*)out, n4);

    // channel 0 accumulation; 2048 blocks * 8 waves = 16384 waves over 200k points
    SoftPoint2Image_splat_kernel<<<2048, 256, 0, stream>>>(p, out, npts);
}